// GCN_33397665694595
// MI455X (gfx1250) — compile-verified
//
#include <hip/hip_runtime.h>
#include <math.h>

typedef __attribute__((ext_vector_type(2))) float v2f;
typedef __attribute__((ext_vector_type(8))) float v8f;

#define B_ 4
#define N_ 50000
#define E_ 1600000
#define H_ 16
#define C_ 10

// ---------------------------------------------------------------- utilities
__global__ void zero_f4(float* __restrict__ p, int n4) {
    int i = blockIdx.x * blockDim.x + threadIdx.x;
    if (i < n4) ((float4*)p)[i] = make_float4(0.f, 0.f, 0.f, 0.f);
}

// x: [B, F=16, N]  ->  h: [B, N, 16]
__global__ void transpose_x(const float* __restrict__ x, float* __restrict__ h) {
    int i = blockIdx.x * blockDim.x + threadIdx.x;      // over B*N
    if (i >= B_ * N_) return;
    int b = i / N_, n = i - b * N_;
    float row[16];
#pragma unroll
    for (int f = 0; f < 16; ++f)
        row[f] = x[((size_t)b * 16 + f) * N_ + n];
    float4* o = (float4*)(h + (size_t)i * 16);
    o[0] = *(float4*)&row[0];  o[1] = *(float4*)&row[4];
    o[2] = *(float4*)&row[8];  o[3] = *(float4*)&row[12];
}

// ------------------------------------------------------------ graph norm
__global__ void deg_scatter(const int* __restrict__ src, float* __restrict__ deg) {
    int e = blockIdx.x * blockDim.x + threadIdx.x;
    if (e < E_) atomicAdd(&deg[src[e]], 1.0f);
}

__global__ void dinv_k(const float* __restrict__ deg, float* __restrict__ dinv) {
    int n = blockIdx.x * blockDim.x + threadIdx.x;
    if (n >= N_) return;
    float d = deg[n];
    dinv[n] = (d > 0.f) ? rsqrtf(fmaxf(d, 1.0f)) : 0.f;
}

__global__ void norm_k(const int* __restrict__ src, const int* __restrict__ dst,
                       const float* __restrict__ dinv, float* __restrict__ nrm) {
    int e = blockIdx.x * blockDim.x + threadIdx.x;
    if (e >= E_) return;
    nrm[e] = -dinv[src[e]] * dinv[dst[e]];
}

// --------------------------------------------- sparse prop: out += norm*z[src]
// grid.y encodes (batch, quarter): b = y>>2, q = y&3 ; each thread = 1 edge, 4 feats
__global__ void prop_scatter(const float* __restrict__ z,
                             const int* __restrict__ src, const int* __restrict__ dst,
                             const float* __restrict__ nrm, float* __restrict__ out) {
    int e = blockIdx.x * blockDim.x + threadIdx.x;
    if (e >= E_) return;
    int b = blockIdx.y >> 2, q = blockIdx.y & 3;
    int s = src[e], d = dst[e];
    float w = nrm[e];
    float4 zv = *(const float4*)(z + ((size_t)b * N_ + s) * 16 + q * 4);
    float* o  = out + ((size_t)b * N_ + d) * 16 + q * 4;
    atomicAdd(o + 0, w * zv.x);
    atomicAdd(o + 1, w * zv.y);
    atomicAdd(o + 2, w * zv.z);
    atomicAdd(o + 3, w * zv.w);
}

// --------------------------------------------------- WMMA combine + ELU
// out = ELU( Tx0*W0^T + Tx1*W1^T + (2P - Tx0)*W2^T + bias )
// One wave per 16-row tile; V_WMMA_F32_16X16X4_F32, K=16 done as 4 chunks of 4.
__global__ __launch_bounds__(256)
void cheb_combine(const float* __restrict__ Tx0, const float* __restrict__ Tx1,
                  const float* __restrict__ P,
                  const float* __restrict__ W,      // [3,16,16] (out, in)
                  const float* __restrict__ bias,   // [16]
                  float* __restrict__ out, int rows) {
    int wave = blockIdx.x * (blockDim.x >> 5) + (threadIdx.x >> 5);
    int lane = threadIdx.x & 31;
    int row0 = wave << 4;
    if (row0 >= rows) return;                // whole-wave uniform: EXEC stays all-1s
    int m   = lane & 15;                     // A: row within tile ; B: output channel
    int hi  = lane >> 4;                     // K half selector
    int off = m * 16 + 2 * hi;

    // Preload B operands: B[kk][n] = W[n][k0+kk]  (mirrored K-striping, fp32 pairs)
    v2f bw[3][4];
#pragma unroll
    for (int k = 0; k < 3; ++k)
#pragma unroll
        for (int kc = 0; kc < 4; ++kc)
            bw[k][kc] = *(const v2f*)(W + k * 256 + off + kc * 4);

    // Accumulator: D[m][n], n = lane&15 -> bias broadcast per output channel
    float bv = bias[m];
    v8f acc;
#pragma unroll
    for (int r = 0; r < 8; ++r) acc[r] = bv;

    const float* a0p = Tx0 + (size_t)row0 * 16 + off;
    const float* a1p = Tx1 + (size_t)row0 * 16 + off;
    const float* app = P   + (size_t)row0 * 16 + off;

#pragma unroll
    for (int kc = 0; kc < 4; ++kc) {
        v2f a = *(const v2f*)(a0p + kc * 4);
        acc = __builtin_amdgcn_wmma_f32_16x16x4_f32(false, a, false, bw[0][kc],
                                                    (short)0, acc, false, false);
    }
#pragma unroll
    for (int kc = 0; kc < 4; ++kc) {
        v2f a = *(const v2f*)(a1p + kc * 4);
        acc = __builtin_amdgcn_wmma_f32_16x16x4_f32(false, a, false, bw[1][kc],
                                                    (short)0, acc, false, false);
    }
#pragma unroll
    for (int kc = 0; kc < 4; ++kc) {   // Tx2 = 2*prop(Tx1) - Tx0, formed on the fly
        v2f p = *(const v2f*)(app + kc * 4);
        v2f a = *(const v2f*)(a0p + kc * 4);
        v2f t;  t.x = 2.f * p.x - a.x;  t.y = 2.f * p.y - a.y;
        acc = __builtin_amdgcn_wmma_f32_16x16x4_f32(false, t, false, bw[2][kc],
                                                    (short)0, acc, false, false);
    }

    // D layout: VGPR r -> row r + 8*hi, col m ; apply ELU and store
#pragma unroll
    for (int r = 0; r < 8; ++r) {
        float v = acc[r];
        v = (v > 0.f) ? v : expm1f(v);
        out[((size_t)(row0 + r + 8 * hi)) * 16 + m] = v;
    }
}

// ------------------------------------------------------------- mean pool
__global__ void pool_k(const float* __restrict__ h, float* __restrict__ g) {
    __shared__ float s[256];
    int b = blockIdx.x, t = threadIdx.x;
    int f = t & 15, ng = t >> 4;              // 16 partial rows per feature
    float acc = 0.f;
    for (int n = ng; n < N_; n += 16)
        acc += h[((size_t)b * N_ + n) * 16 + f];
    s[t] = acc;
    __syncthreads();
    for (int st = 128; st >= 16; st >>= 1) {
        if (t < st) s[t] += s[t + st];
        __syncthreads();
    }
    if (t < 16) g[b * 16 + t] = s[t] * (1.0f / (float)N_);
}

// ---------------------------------------------------- linear + log_softmax
__global__ void head_k(const float* __restrict__ g, const float* __restrict__ lw,
                       const float* __restrict__ lb, float* __restrict__ out) {
    __shared__ float lg[B_][C_];
    int t = threadIdx.x;
    if (t < B_ * C_) {
        int b = t / C_, c = t - b * C_;
        float a = lb[c];
#pragma unroll
        for (int h = 0; h < 16; ++h) a += g[b * 16 + h] * lw[c * 16 + h];
        lg[b][c] = a;
    }
    __syncthreads();
    if (t < B_ * C_) {
        int b = t / C_, c = t - b * C_;
        float mx = -1e30f;
        for (int j = 0; j < C_; ++j) mx = fmaxf(mx, lg[b][j]);
        float se = 0.f;
        for (int j = 0; j < C_; ++j) se += expf(lg[b][j] - mx);
        out[b * C_ + c] = lg[b][c] - mx - logf(se);
    }
}

// ---------------------------------------------------------------- launcher
extern "C" void kernel_launch(void* const* d_in, const int* in_sizes, int n_in,
                              void* d_out, int out_size, void* d_ws, size_t ws_size,
                              hipStream_t stream) {
    const float* x  = (const float*)d_in[0];
    const int*   ei = (const int*)d_in[1];          // [1,2,E] -> src | dst
    const float* Wk[3] = {(const float*)d_in[2], (const float*)d_in[4], (const float*)d_in[6]};
    const float* bk[3] = {(const float*)d_in[3], (const float*)d_in[5], (const float*)d_in[7]};
    const float* lw = (const float*)d_in[8];
    const float* lb = (const float*)d_in[9];
    const int* src = ei;
    const int* dst = ei + E_;
    float* out = (float*)d_out;

    // workspace carve-up (~58 MB of f32)
    float* ws = (float*)d_ws;
    size_t o = 0;
    float* deg  = ws + o;  o += N_;
    float* dinv = ws + o;  o += N_;
    float* nrm  = ws + o;  o += E_;
    const size_t NB = (size_t)B_ * N_ * 16;         // 3.2M floats per node buffer
    float* bufA = ws + o;  o += NB;
    float* bufB = ws + o;  o += NB;
    float* bufC = ws + o;  o += NB;
    float* bufD = ws + o;  o += NB;
    float* g    = ws + o;  o += 64;
    (void)ws_size; (void)in_sizes; (void)n_in; (void)out_size;

    const int T = 256;
    // graph normalization
    zero_f4<<<(N_ / 4 + T - 1) / T, T, 0, stream>>>(deg, N_ / 4);
    deg_scatter<<<(E_ + T - 1) / T, T, 0, stream>>>(src, deg);
    dinv_k<<<(N_ + T - 1) / T, T, 0, stream>>>(deg, dinv);
    norm_k<<<(E_ + T - 1) / T, T, 0, stream>>>(src, dst, dinv, nrm);

    // x[B,16,N] -> h0[B,N,16]
    transpose_x<<<(B_ * N_ + T - 1) / T, T, 0, stream>>>(x, bufA);

    float* cur = bufA;
    float* nxt = bufD;
    const int rows   = B_ * N_;
    const int tiles  = rows / 16;
    const int cblks  = (tiles + 7) / 8;             // 8 waves/block
    dim3 pgrid((E_ + T - 1) / T, B_ * 4);

    for (int l = 0; l < 3; ++l) {
        zero_f4<<<((int)(NB / 4) + T - 1) / T, T, 0, stream>>>(bufB, (int)(NB / 4));
        prop_scatter<<<pgrid, T, 0, stream>>>(cur, src, dst, nrm, bufB);    // Tx1
        zero_f4<<<((int)(NB / 4) + T - 1) / T, T, 0, stream>>>(bufC, (int)(NB / 4));
        prop_scatter<<<pgrid, T, 0, stream>>>(bufB, src, dst, nrm, bufC);   // prop(Tx1)
        cheb_combine<<<cblks, T, 0, stream>>>(cur, bufB, bufC, Wk[l], bk[l], nxt, rows);
        float* tmp = cur; cur = nxt; nxt = tmp;
    }

    pool_k<<<B_, 256, 0, stream>>>(cur, g);
    head_k<<<1, 64, 0, stream>>>(g, lw, lb, out);
}